// GraphSAGEEncoder_79096117723243
// MI455X (gfx1250) — compile-verified
//
#include <hip/hip_runtime.h>
#include <hip/hip_bf16.h>

// GraphSAGE 2-layer encoder for MI455X (gfx1250, wave32).
// Pipeline: zero -> edge scatter (f32 global atomics, L2-resident) ->
//           fused WMMA-f32 GEMM ([mean|x] @ [Wl|Wr]^T + b, relu) -> repeat.
// Workspace layout: sum[N*256] | h[N*256] | deg[N]  (~205.2 MB)

typedef float v2f __attribute__((ext_vector_type(2)));
typedef float v8f __attribute__((ext_vector_type(8)));

#define DIN 256
#define TILE_M 16

// ---------------------------------------------------------------- zeroing
__global__ void zero4_kernel(float4* __restrict__ p, long n4) {
    long i = (long)blockIdx.x * blockDim.x + threadIdx.x;
    long stride = (long)gridDim.x * blockDim.x;
    float4 z = make_float4(0.f, 0.f, 0.f, 0.f);
    for (; i < n4; i += stride) p[i] = z;
}

// ---------------------------------------------------------------- scatter
// One block (64 threads = 2 waves) per edge: sum[dst] += feat[src]; deg[dst]++.
__global__ void scatter_kernel(const float* __restrict__ feat,
                               const int* __restrict__ src,
                               const int* __restrict__ dst,
                               float* __restrict__ sum,
                               float* __restrict__ deg,
                               int countDeg) {
    int e = blockIdx.x;
    int s = src[e];
    int d = dst[e];
    float4 v = ((const float4*)(feat + (size_t)s * DIN))[threadIdx.x];
    float* o = sum + (size_t)d * DIN + threadIdx.x * 4;
    unsafeAtomicAdd(o + 0, v.x);
    unsafeAtomicAdd(o + 1, v.y);
    unsafeAtomicAdd(o + 2, v.z);
    unsafeAtomicAdd(o + 3, v.w);
    if (countDeg && threadIdx.x == 0) unsafeAtomicAdd(&deg[d], 1.0f);
}

// ---------------------------------------------------------------- fused GEMM
// out[n,o] = sum_k mean[n,k]*Wl[o,k] + sum_k x[n,k]*Wr[o,k] + b[o]  (opt relu)
// Block = 256 threads (8 waves), 16 nodes/block. K = 2*DIN = 512 staged in LDS.
template <int OUT, bool RELU>
__global__ __launch_bounds__(256) void sage_gemm_kernel(
    const float* __restrict__ sum, const float* __restrict__ deg,
    const float* __restrict__ x,
    const float* __restrict__ Wl, const float* __restrict__ bias,
    const float* __restrict__ Wr,
    float* __restrict__ out) {

    __shared__ float ldsA[TILE_M * 2 * DIN];   // [16 rows][512] = 32 KB
    __shared__ float invdeg[TILE_M];

    const int tid = threadIdx.x;
    const int nodeBase = blockIdx.x * TILE_M;

    if (tid < TILE_M) {
        float dg = deg[nodeBase + tid];
        invdeg[tid] = 1.0f / fmaxf(dg, 1.0f);
    }
    __syncthreads();

    // Stage A = [mean | x] into LDS (float4 per iteration; 8 iters/thread)
    for (int i = tid; i < TILE_M * (2 * DIN) / 4; i += 256) {
        int row = i >> 7;            // 128 float4 per row
        int col = (i & 127) * 4;
        float4 v;
        if (col < DIN) {
            v = *(const float4*)&sum[(size_t)(nodeBase + row) * DIN + col];
            float sc = invdeg[row];
            v.x *= sc; v.y *= sc; v.z *= sc; v.w *= sc;
        } else {
            v = *(const float4*)&x[(size_t)(nodeBase + row) * DIN + (col - DIN)];
        }
        *(float4*)&ldsA[row * (2 * DIN) + col] = v;
    }
    __syncthreads();

    const int lane   = tid & 31;
    const int wave   = tid >> 5;
    const int lmod   = lane & 15;        // M-row (A) / N-col (B,C)
    const int rowSel = lane >> 4;        // selects K pair {0,1} vs {2,3}

    const float* aBase = &ldsA[lmod * (2 * DIN) + rowSel * 2];

    for (int ot = wave; ot < OUT / 16; ot += 8) {
        const int obase = ot * 16;
        const float* bl = Wl + (size_t)(obase + lmod) * DIN + rowSel * 2;
        const float* br = Wr + (size_t)(obase + lmod) * DIN + rowSel * 2;

        v8f c = {0.f, 0.f, 0.f, 0.f, 0.f, 0.f, 0.f, 0.f};

        // K half 1: mean part against Wl
#pragma unroll 8
        for (int k = 0; k < DIN; k += 4) {
            v2f a = *(const v2f*)(aBase + k);
            v2f b = *(const v2f*)(bl + k);
            c = __builtin_amdgcn_wmma_f32_16x16x4_f32(
                    false, a, false, b, (short)0, c, false, false);
        }
        // K half 2: x part against Wr
#pragma unroll 8
        for (int k = 0; k < DIN; k += 4) {
            v2f a = *(const v2f*)(aBase + DIN + k);
            v2f b = *(const v2f*)(br + k);
            c = __builtin_amdgcn_wmma_f32_16x16x4_f32(
                    false, a, false, b, (short)0, c, false, false);
        }

        float bv = bias[obase + lmod];
#pragma unroll
        for (int v = 0; v < 8; ++v) {
            int row = v + rowSel * 8;
            float val = c[v] + bv;
            if (RELU) val = fmaxf(val, 0.0f);
            out[(size_t)(nodeBase + row) * OUT + obase + lmod] = val;
        }
    }
}

// ---------------------------------------------------------------- launch
extern "C" void kernel_launch(void* const* d_in, const int* in_sizes, int n_in,
                              void* d_out, int out_size, void* d_ws, size_t ws_size,
                              hipStream_t stream) {
    const float* x   = (const float*)d_in[0];
    const int*   ei  = (const int*)d_in[1];
    const float* W1l = (const float*)d_in[2];
    const float* b1  = (const float*)d_in[3];
    const float* W1r = (const float*)d_in[4];
    const float* W2l = (const float*)d_in[5];
    const float* b2  = (const float*)d_in[6];
    const float* W2r = (const float*)d_in[7];

    const int N = in_sizes[0] / DIN;      // 100000
    const int E = in_sizes[1] / 2;        // 1600000

    const int* src = ei;
    const int* dst = ei + E;

    float* sum = (float*)d_ws;
    float* h   = sum + (size_t)N * DIN;
    float* deg = h   + (size_t)N * DIN;
    float* out = (float*)d_out;

    const long sumN4 = (long)N * DIN / 4;
    const long degN4 = (long)N / 4;

    // Layer 1
    zero4_kernel<<<4096, 256, 0, stream>>>((float4*)sum, sumN4);
    zero4_kernel<<<256, 256, 0, stream>>>((float4*)deg, degN4);
    scatter_kernel<<<E, 64, 0, stream>>>(x, src, dst, sum, deg, 1);
    sage_gemm_kernel<256, true><<<N / TILE_M, 256, 0, stream>>>(
        sum, deg, x, W1l, b1, W1r, h);

    // Layer 2 (reuse sum buffer; degree is identical)
    zero4_kernel<<<4096, 256, 0, stream>>>((float4*)sum, sumN4);
    scatter_kernel<<<E, 64, 0, stream>>>(h, src, dst, sum, deg, 0);
    sage_gemm_kernel<128, false><<<N / TILE_M, 256, 0, stream>>>(
        sum, deg, h, W2l, b2, W2r, out);
}